// OnlineNorm_11982958756550
// MI455X (gfx1250) — compile-verified
//
#include <hip/hip_runtime.h>
#include <stdint.h>

typedef float v2f __attribute__((ext_vector_type(2)));
typedef float v8f __attribute__((ext_vector_type(8)));

#define EPS_N 1e-12f
constexpr int Bsz  = 16;
constexpr int Tlen = 3000;
constexpr int Fdim = 513;
constexpr int FT   = 33;    // ceil(513/16)
constexpr int NSEG = 5;
constexpr int SEGLEN = 608; // multiple of 16; 5*608 >= 3000
constexpr int WARM = 208;   // multiple of 16; (1-a)^208 ~ 3e-15 << fp32 eps
constexpr int ROWP = 20;    // padded LDS row stride (floats) -> conflict-free column reads

__device__ __forceinline__ float cpowi(float c, int n) {
    float p = 1.f;
    for (int k = 0; k < n; ++k) p *= c;
    return p;
}

// Truncate generic pointer to wave-relative LDS byte offset (low 32 bits of
// the LDS aperture address are the offset per CDNA5 aperture layout).
__device__ __forceinline__ uint32_t lds_off(const float* p) {
    return (uint32_t)(uint64_t)(uintptr_t)p;
}

// Async-copy one 16x16 fp32 tile (time rows tt..tt+15, channel cols fc-tile)
// into LDS. 8 x global_load_async_to_lds_b32: 32 lanes * 8 = 256 dwords.
__device__ __forceinline__ void async_tile_load(const float* gbase, int tt, int fc,
                                                const float* ldsbuf, int lane) {
    const int col = lane & 15;
    const int hi  = lane >> 4;
    // our own ds reads of this buffer must complete before the async engine
    // overwrites it
    asm volatile("s_wait_dscnt 0" ::: "memory");
#pragma unroll
    for (int q = 0; q < 8; ++q) {
        int trow = hi + 2 * q;
        int t = tt + trow;
        t = (t < Tlen) ? t : (Tlen - 1);                  // clamp tail loads
        uint64_t ga = (uint64_t)(uintptr_t)(gbase + (size_t)t * Fdim + fc);
        uint32_t la = lds_off(ldsbuf + trow * ROWP + col);
        asm volatile("global_load_async_to_lds_b32 %0, %1, off"
                     :: "v"(la), "v"(ga) : "memory");
    }
}

__global__ void __launch_bounds__(32)
onlinenorm_wmma_kernel(const float* __restrict__ x,
                       const float* __restrict__ rmean,
                       const float* __restrict__ rvar,
                       const float* __restrict__ alphap,
                       float* __restrict__ out) {
    __shared__ float lds[2][16 * ROWP];

    const int lane  = threadIdx.x;
    const int col   = lane & 15;   // C/B-layout column (channel within tile)
    const int hi    = lane >> 4;   // which half of the wave
    const int rbase = hi * 8;      // C-layout row base for this lane

    int blk = blockIdx.x;
    const int seg = blk % NSEG; blk /= NSEG;
    const int ft  = blk % FT;   blk /= FT;
    const int b   = blk;

    const float a = alphap[0];
    const float c = 1.0f - a;

    const int f  = ft * 16 + col;
    const int fc = (f < Fdim) ? f : (Fdim - 1);           // clamp channel for loads

    const int t0   = seg * SEGLEN;
    const int tend = (t0 + SEGLEN < Tlen) ? (t0 + SEGLEN) : Tlen;
    const int ts   = (seg == 0) ? 0 : (t0 - WARM);        // warm-up start
    const int nch  = (tend - ts + 15) >> 4;               // 16-step tiles

    // Constant A operands: lower-triangular decay matrices, WMMA A layout
    // (A 16x4 f32: lanes hold M=lane&15; VGPR0 = K0/K2, VGPR1 = K1/K3).
    v2f Am[4], Av[4];
#pragma unroll
    for (int kb = 0; kb < 4; ++kb) {
        const int row = lane & 15;
        const int jx = kb * 4 + (hi ? 2 : 0);
        const int jy = jx + 1;
        const float ex = (jx <= row) ? cpowi(c, row - jx) : 0.f;
        const float ey = (jy <= row) ? cpowi(c, row - jy) : 0.f;
        Am[kb].x = a * ex; Am[kb].y = a * ey;   // mean: a * c^(i-j)
        Av[kb].x = ex;     Av[kb].y = ey;       // var :     c^(i-j)
    }
    float dm[8];                                // init-state decay c^(i+1), C layout
#pragma unroll
    for (int v = 0; v < 8; ++v) dm[v] = cpowi(c, rbase + v + 1);

    float m_prev = rmean[fc];
    float v_prev = rvar[fc];

    const float* gb = x + (size_t)b * Tlen * Fdim;

    async_tile_load(gb, ts, fc, lds[0], lane);

    int tt = ts;
    for (int ci = 0; ci < nch; ++ci) {
        const int cur = ci & 1;
        if (ci + 1 < nch) {
            async_tile_load(gb, tt + 16, fc, lds[cur ^ 1], lane);
            asm volatile("s_wait_asynccnt 8" ::: "memory");   // current tile done
        } else {
            asm volatile("s_wait_asynccnt 0" ::: "memory");
        }
        const float* tile = lds[cur];

        // ---- mean tile: M = Lm @ X + d (x) m_prev : 4 chained f32 WMMAs ----
        v8f M;
#pragma unroll
        for (int v = 0; v < 8; ++v) M[v] = dm[v] * m_prev;
#pragma unroll
        for (int kb = 0; kb < 4; ++kb) {
            const int r0 = kb * 4 + hi * 2;     // B layout: VGPR0 = K0/K2, VGPR1 = K1/K3
            v2f Bx;
            Bx.x = tile[r0 * ROWP + col];
            Bx.y = tile[(r0 + 1) * ROWP + col];
            M = __builtin_amdgcn_wmma_f32_16x16x4_f32(false, Am[kb], false, Bx,
                                                      (short)0, M, false, false);
        }

        // x in C layout for residual + output
        float xc[8];
#pragma unroll
        for (int v = 0; v < 8; ++v) xc[v] = tile[(rbase + v) * ROWP + col];

        // residual r = a*(x-m)^2 (C layout) + cross-half exchange for B layout
        float r[8], ro[8];
#pragma unroll
        for (int v = 0; v < 8; ++v) { float d0 = xc[v] - M[v]; r[v] = a * d0 * d0; }
#pragma unroll
        for (int v = 0; v < 8; ++v) ro[v] = __shfl_xor(r[v], 16, 32);

        // ---- var tile: V = Lv @ R + d (x) v_prev ----
        v8f V;
#pragma unroll
        for (int v = 0; v < 8; ++v) V[v] = dm[v] * v_prev;
        {
            v2f B0; B0.x = hi ? ro[2] : r[0]; B0.y = hi ? ro[3] : r[1];
            V = __builtin_amdgcn_wmma_f32_16x16x4_f32(false, Av[0], false, B0, (short)0, V, false, false);
            v2f B1; B1.x = hi ? ro[6] : r[4]; B1.y = hi ? ro[7] : r[5];
            V = __builtin_amdgcn_wmma_f32_16x16x4_f32(false, Av[1], false, B1, (short)0, V, false, false);
            v2f B2; B2.x = hi ? r[2] : ro[0]; B2.y = hi ? r[3] : ro[1];
            V = __builtin_amdgcn_wmma_f32_16x16x4_f32(false, Av[2], false, B2, (short)0, V, false, false);
            v2f B3; B3.x = hi ? r[6] : ro[4]; B3.y = hi ? r[7] : ro[5];
            V = __builtin_amdgcn_wmma_f32_16x16x4_f32(false, Av[3], false, B3, (short)0, V, false, false);
        }

        // carry = row 15 of M and V (held in lane 16+col, VGPR 7)
        m_prev = __shfl(M[7], 16 + col, 32);
        v_prev = __shfl(V[7], 16 + col, 32);

        // store compute-region rows only (skip warm-up tiles, clamp T and F)
        if (tt >= t0 && f < Fdim) {
#pragma unroll
            for (int v = 0; v < 8; ++v) {
                const int t = tt + rbase + v;
                if (t < tend)
                    out[((size_t)b * Tlen + t) * Fdim + f] =
                        (xc[v] - M[v]) / (4.0f * V[v] + EPS_N);
            }
        }
        tt += 16;
    }
}

extern "C" void kernel_launch(void* const* d_in, const int* in_sizes, int n_in,
                              void* d_out, int out_size, void* d_ws, size_t ws_size,
                              hipStream_t stream) {
    const float* x  = (const float*)d_in[0];
    const float* rm = (const float*)d_in[1];
    const float* rv = (const float*)d_in[2];
    const float* al = (const float*)d_in[3];
    float* out = (float*)d_out;
    (void)in_sizes; (void)n_in; (void)out_size; (void)d_ws; (void)ws_size;

    dim3 grid(Bsz * FT * NSEG);   // 16 * 33 * 5 = 2640 single-wave blocks
    onlinenorm_wmma_kernel<<<grid, 32, 0, stream>>>(x, rm, rv, al, out);
}